// TestNet_38620345925971
// MI455X (gfx1250) — compile-verified
//
#include <hip/hip_runtime.h>

// -------- CDNA5 WMMA types --------
typedef __attribute__((ext_vector_type(2))) float v2f;
typedef __attribute__((ext_vector_type(8))) float v8f;

#define SA 192      // sinogram angles
#define SD 576      // detector count
#define IW 384      // image width
#define TX 64       // conv tile width  (sino_pred)
#define TY 8        // conv tile height (sino_pred)
#define HW (TX + 4) // 68: h tile width incl. halo
#define HH (TY + 4) // 12: h tile height incl. halo
#define SCOLS (TX + 8) // 72: staged sino cols
#define SROWS (TY + 8) // 16: staged sino rows

// ======================================================================
// Fused conv1 (1->16, 5x5, pad 2) -> conv2 (16->1, 5x5, pad 2).
// Step 0: async-copy the (zero-padded) sino window into LDS.
// Stage 1: h tile (16ch x HH x HW, halo included) via f32 WMMA into LDS.
// Stage 2: conv2 as K=400 WMMA reduction (w2 replicated across M rows).
// ======================================================================
__global__ __launch_bounds__(256) void
fanbeam_conv12_kernel(const float* __restrict__ sino,
                      const float* __restrict__ w1, const float* __restrict__ b1,
                      const float* __restrict__ w2, const float* __restrict__ b2,
                      float* __restrict__ sino_pred)
{
    __shared__ float h_lds[16 * HH * HW];      // [ch][row][col], 52224 B
    __shared__ float sino_lds[SROWS * SCOLS];  // zero-padded sino window, 4608 B
    __shared__ float w2_lds[400];

    const int tid    = threadIdx.x;
    const int lane   = tid & 31;
    const int wv     = tid >> 5;        // wave 0..7
    const int halfHi = lane >> 4;       // 0: lanes 0-15, 1: lanes 16-31
    const int lm     = lane & 15;

    const int bx = (blockIdx.x % (SD / TX)) * TX; // 0..512
    const int by = (blockIdx.x / (SD / TX)) * TY; // 0..184

    for (int i = tid; i < 400; i += 256) w2_lds[i] = w2[i];

    // ---- Step 0: stage sino window [by-4, by+TY+4) x [bx-4, bx+TX+4) ----
    // In-bounds cells: CDNA5 async global->LDS copy (ASYNCcnt-tracked).
    // Out-of-bounds cells: zero-filled via DS store (disjoint addresses).
    for (int i = tid; i < SROWS * SCOLS; i += 256) {
        int row = i / SCOLS, c = i - row * SCOLS;
        int gy = by - 4 + row, gx = bx - 4 + c;
        bool inb = (gy >= 0) && (gy < SA) && (gx >= 0) && (gx < SD);
        if (inb) {
            // GV mode: vdst = LDS byte offset, vaddr = 64-bit global address
            unsigned lds_byte = (unsigned)(unsigned long long)(&sino_lds[i]);
            const float* gptr = sino + gy * SD + gx;
            asm volatile("global_load_async_to_lds_b32 %0, %1, off"
                         :: "v"(lds_byte), "v"(gptr) : "memory");
        } else {
            sino_lds[i] = 0.0f;
        }
    }
    asm volatile("s_wait_asynccnt 0" ::: "memory");

    // Preload A = w1 [16ch x 25] in WMMA 16x4 f32 A-layout, 7 K-chunks.
    // lane(0-15): {A[M=lm,kb], A[M=lm,kb+1]}, lane(16-31): {A[M=lm,kb+2], A[M=lm,kb+3]}
    v2f aw1[7];
    #pragma unroll
    for (int ks = 0; ks < 7; ++ks) {
        int kb = ks * 4 + halfHi * 2;
        aw1[ks][0] = (kb     < 25) ? w1[lm * 25 + kb]     : 0.0f;
        aw1[ks][1] = (kb + 1 < 25) ? w1[lm * 25 + kb + 1] : 0.0f;
    }
    float b1v[8];
    #pragma unroll
    for (int r = 0; r < 8; ++r) b1v[r] = b1[r + halfHi * 8];
    const float b2v = b2[0];

    __syncthreads();

    // ---- Stage 1: conv1 into LDS (60 subtiles of 16 cols x 1 row) ----
    for (int t = wv; t < 5 * HH; t += 8) {
        const int r    = t / 5;               // h row 0..11
        const int col  = (t % 5) * 16 + lm;   // h col 0..79 (valid < HW)
        const int lcol = min(col, HW - 1);    // clamp dead lanes into staged range
        const int hy   = by - 2 + r;
        const int hx   = bx - 2 + col;
        const bool hvalid = (hy >= 0) & (hy < SA) & (hx >= 0) & (hx < SD) & (col < HW);

        v8f acc = {};
        #pragma unroll
        for (int ks = 0; ks < 7; ++ks) {
            int kb = ks * 4 + halfHi * 2;
            v2f bb;
            #pragma unroll
            for (int j = 0; j < 2; ++j) {
                int k  = kb + j;
                int kc = min(k, 24);                  // keep LDS address in range
                int ky = kc / 5, kx = kc - ky * 5;
                float v = sino_lds[(r + ky) * SCOLS + lcol + kx]; // unconditional ds_load
                bb[j] = (k < 25) ? v : 0.0f;          // cndmask, no branch
            }
            acc = __builtin_amdgcn_wmma_f32_16x16x4_f32(
                false, aw1[ks], false, bb, (short)0, acc, false, false);
        }
        // D layout: VGPR r -> ch = r + 8*halfHi, pixel = lm
        #pragma unroll
        for (int rr = 0; rr < 8; ++rr) {
            int ch = rr + halfHi * 8;
            float val = hvalid ? (acc[rr] + b1v[rr]) : 0.0f;
            if (col < HW)
                h_lds[(ch * HH + r) * HW + col] = val;
        }
    }
    __syncthreads();

    // ---- Stage 2: conv2 from LDS, 32 output subtiles of 16 pixels ----
    for (int t = wv; t < (TX / 16) * TY; t += 8) {
        const int oy = t >> 2;         // 0..7
        const int ox = (t & 3) * 16;   // 0,16,32,48
        v8f acc = {};
        for (int kb4 = 0; kb4 < 400; kb4 += 4) {
            int kb = kb4 + halfHi * 2;
            v2f av, bv;
            #pragma unroll
            for (int j = 0; j < 2; ++j) {
                int k  = kb + j;             // (c, ky, kx) flattened, c-major
                int c  = k / 25;
                int s  = k - c * 25;
                int ky = s / 5;
                int kx = s - ky * 5;
                av[j] = w2_lds[k];                                     // replicated over M
                bv[j] = h_lds[(c * HH + (oy + ky)) * HW + (ox + lm + kx)];
            }
            acc = __builtin_amdgcn_wmma_f32_16x16x4_f32(
                false, av, false, bv, (short)0, acc, false, false);
        }
        // Every D row equals dot(w2, patch_n); row M=0 lives in acc[0], lanes 0-15.
        if (lane < 16) {
            int y = by + oy, x = bx + ox + lm;
            sino_pred[y * SD + x] = acc[0] + b2v;
        }
    }
}

// ======================================================================
// Dual fan-beam backprojection: one s-coordinate computation feeds
// gathers from both sino and sino_pred (both L2-resident).
// ======================================================================
__global__ __launch_bounds__(256) void
fanbeam_backproj2_kernel(const float* __restrict__ sino,
                         const float* __restrict__ sino_pred,
                         const float* __restrict__ angles,
                         float* __restrict__ img,
                         float* __restrict__ img_sino)
{
    __shared__ float cs[SA], sn[SA];
    const int tid = threadIdx.x;
    if (tid < SA) {
        float a = angles[tid];
        cs[tid] = cosf(a);
        sn[tid] = sinf(a);
    }
    __syncthreads();

    const int pix = blockIdx.x * 256 + tid;
    if (pix >= IW * IW) return;
    const int x = pix % IW, y = pix / IW;
    const float X = (float)x - (float)(IW - 1) * 0.5f;
    const float Y = (float)y - (float)(IW - 1) * 0.5f;

    float acc0 = 0.0f, acc1 = 0.0f;
    for (int a = 0; a < SA; ++a) {
        float cb = cs[a], sb = sn[a];
        float px = cb * X + sb * Y;
        float py = cb * Y - sb * X;
        float s  = (float)SD * px / ((float)SD + py) + (float)(SD - 1) * 0.5f;
        float i0f = floorf(s);
        float fr  = s - i0f;
        int i0 = (int)i0f;
        int i1 = i0 + 1;
        float w0 = (i0 >= 0 && i0 < SD) ? (1.0f - fr) : 0.0f;
        float w1 = (i1 >= 0 && i1 < SD) ? fr : 0.0f;
        int i0c = min(max(i0, 0), SD - 1);
        int i1c = min(max(i1, 0), SD - 1);
        const float* r0 = sino      + a * SD;
        const float* r1 = sino_pred + a * SD;
        acc0 += r0[i0c] * w0 + r0[i1c] * w1;
        acc1 += r1[i0c] * w0 + r1[i1c] * w1;
    }
    img[pix]      = acc0;
    img_sino[pix] = acc1;
}

// ======================================================================
// conv3: 2->1 channels, 3x3, pad 1 on 384x384 (channels: img, img_sino)
// ======================================================================
__global__ __launch_bounds__(256) void
fanbeam_conv3_kernel(const float* __restrict__ img,
                     const float* __restrict__ img_sino,
                     const float* __restrict__ w3,
                     const float* __restrict__ b3,
                     float* __restrict__ img_pred)
{
    const int pix = blockIdx.x * 256 + threadIdx.x;
    if (pix >= IW * IW) return;
    const int x = pix % IW, y = pix / IW;

    float acc = b3[0];
    #pragma unroll
    for (int c = 0; c < 2; ++c) {
        const float* src = c ? img_sino : img;
        #pragma unroll
        for (int ky = 0; ky < 3; ++ky) {
            int sy = y + ky - 1;
            if (sy < 0 || sy >= IW) continue;
            #pragma unroll
            for (int kx = 0; kx < 3; ++kx) {
                int sx = x + kx - 1;
                if (sx < 0 || sx >= IW) continue;
                acc += w3[(c * 3 + ky) * 3 + kx] * src[sy * IW + sx];
            }
        }
    }
    img_pred[pix] = acc;
}

extern "C" void kernel_launch(void* const* d_in, const int* in_sizes, int n_in,
                              void* d_out, int out_size, void* d_ws, size_t ws_size,
                              hipStream_t stream)
{
    const float* sino   = (const float*)d_in[0];
    const float* angles = (const float*)d_in[1];
    const float* w1     = (const float*)d_in[2];
    const float* b1     = (const float*)d_in[3];
    const float* w2     = (const float*)d_in[4];
    const float* b2     = (const float*)d_in[5];
    const float* w3     = (const float*)d_in[6];
    const float* b3     = (const float*)d_in[7];

    float* out       = (float*)d_out;
    float* sino_pred = out;                       // [1,1,192,576]
    float* img_sino  = sino_pred + SA * SD;       // [1,1,384,384]
    float* img_pred  = img_sino + IW * IW;        // [1,1,384,384]
    float* img       = (float*)d_ws;              // scratch: [1,1,384,384]

    const int conv_blocks = (SD / TX) * (SA / TY);          // 9*24 = 216
    const int pix_blocks  = (IW * IW + 255) / 256;          // 576

    fanbeam_conv12_kernel<<<conv_blocks, 256, 0, stream>>>(
        sino, w1, b1, w2, b2, sino_pred);
    fanbeam_backproj2_kernel<<<pix_blocks, 256, 0, stream>>>(
        sino, sino_pred, angles, img, img_sino);
    fanbeam_conv3_kernel<<<pix_blocks, 256, 0, stream>>>(
        img, img_sino, w3, b3, img_pred);
}